// SoftRas_19138374271069
// MI455X (gfx1250) — compile-verified
//
#include <hip/hip_runtime.h>
#include <math.h>

// SoftRas forward for MI455X (gfx1250).
// VALU/transcendental-bound (≈1.6 GFLOP, ≈200KB total data) — HBM irrelevant.
// Strategy: LDS-resident face table staged via CDNA5 async global->LDS
// (ASYNCcnt + s_wait_asynccnt), broadcast ds_load_b128 reads in the hot loop,
// 2 pixels/thread for ILP across the exp/log1p TRANS32 chains, deterministic
// chunked accumulation (no float atomics -> bitwise reproducible replays).

#define IMGW    96
#define NPIX    (IMGW * IMGW)      // 9216
#define PXT     512                // pixels per block (2 per thread)
#define FCHUNK  256                // faces per chunk staged in LDS

#if defined(__AMDGCN__) && __has_builtin(__builtin_amdgcn_global_load_async_to_lds_b128)
#define HAS_ASYNC_LDS 1
#else
#define HAS_ASYNC_LDS 0
#endif

typedef int v4i __attribute__((vector_size(16)));
typedef __attribute__((address_space(1))) v4i* g_v4i_p;
typedef __attribute__((address_space(3))) v4i* l_v4i_p;

__device__ __forceinline__ void stage_f4(const float4* __restrict__ g,
                                         float4* __restrict__ l, int idx) {
#if HAS_ASYNC_LDS
  __builtin_amdgcn_global_load_async_to_lds_b128(
      (g_v4i_p)(g + idx),
      (l_v4i_p)(l + idx),
      /*offset=*/0, /*cpol=*/0);
#else
  l[idx] = g[idx];
#endif
}

__device__ __forceinline__ void wait_async0() {
#if HAS_ASYNC_LDS
#if __has_builtin(__builtin_amdgcn_s_wait_asynccnt)
  __builtin_amdgcn_s_wait_asynccnt(0);
#else
  asm volatile("s_wait_asynccnt 0x0" ::: "memory");
#endif
#endif
}

// ---------------------------------------------------------------------------
// Kernel 1: per-vertex projection. grid = (ceil(V/256), B) -> no runtime
// div/mod. Look-at Rl is identity for the reference's fixed eye/up.
// ---------------------------------------------------------------------------
__global__ __launch_bounds__(256) void vert_kernel(
    const float* __restrict__ verts, const float* __restrict__ cams,
    float* __restrict__ xy, int V) {
  const int n = blockIdx.x * blockDim.x + threadIdx.x;
  if (n >= V) return;
  const int b = blockIdx.y;
  const float* c = cams + (size_t)b * 7;
  float s = c[0], tx = c[1], ty = c[2];
  float qw = c[3], qx = c[4], qy = c[5], qz = c[6];
  float qn = sqrtf(qw * qw + qx * qx + qy * qy + qz * qz) + 1e-8f;
  float inv_qn = 1.0f / qn;
  qw *= inv_qn; qx *= inv_qn; qy *= inv_qn; qz *= inv_qn;

  float r00 = 1.f - 2.f * (qy * qy + qz * qz);
  float r01 = 2.f * (qx * qy - qw * qz);
  float r02 = 2.f * (qx * qz + qw * qy);
  float r10 = 2.f * (qx * qy + qw * qz);
  float r11 = 1.f - 2.f * (qx * qx + qz * qz);
  float r12 = 2.f * (qy * qz - qw * qx);
  float r20 = 2.f * (qx * qz - qw * qy);
  float r21 = 2.f * (qy * qz + qw * qx);
  float r22 = 1.f - 2.f * (qx * qx + qy * qy);

  const size_t i = (size_t)b * V + n;
  const float* vp = verts + i * 3;
  float vx = vp[0], vy = vp[1], vz = vp[2];
  float X = r00 * vx + r01 * vy + r02 * vz;
  float Y = r10 * vx + r11 * vy + r12 * vz;
  float Z = r20 * vx + r21 * vy + r22 * vz;

  float px = s * X + tx;
  float py = -(s * Y + ty);               // * [1,-1,1]
  float pz = s * Z + 5.0f + 2.732f;       // + OFFSET_Z, - eye.z  (Rl == I)
  const float kWidth = 0.57735026918962576f;  // tan(30 deg)
  float inv = 1.0f / (pz * kWidth);
  xy[2 * i]     = px * inv;
  xy[2 * i + 1] = py * inv;
}

// ---------------------------------------------------------------------------
// Kernel 2: per-face record, 16 floats (64B, b128-friendly). grid=(F/256, B).
// [a0x a0y e0x e0y | a1x a1y e1x e1y | a2x a2y e2x e2y | iee0 iee1 iee2 pad]
// ---------------------------------------------------------------------------
__global__ __launch_bounds__(256) void face_kernel(
    const int* __restrict__ faces, const float* __restrict__ xy,
    float* __restrict__ rec, int V, int F) {
  const int f = blockIdx.x * blockDim.x + threadIdx.x;
  if (f >= F) return;
  const int b = blockIdx.y;
  const size_t i = (size_t)b * F + f;
  const int* fp = faces + i * 3;
  float ax[3], ay[3];
#pragma unroll
  for (int k = 0; k < 3; ++k) {
    int idx = fp[k];
    ax[k] = xy[2 * ((size_t)b * V + idx)];
    ay[k] = xy[2 * ((size_t)b * V + idx) + 1];
  }
  float* r = rec + i * 16;
  float iee[3];
#pragma unroll
  for (int k = 0; k < 3; ++k) {
    int k1 = (k + 1) % 3;
    float ex = ax[k1] - ax[k];
    float ey = ay[k1] - ay[k];
    r[4 * k + 0] = ax[k];
    r[4 * k + 1] = ay[k];
    r[4 * k + 2] = ex;
    r[4 * k + 3] = ey;
    iee[k] = 1.0f / (ex * ex + ey * ey + 1e-12f);
  }
  r[12] = iee[0]; r[13] = iee[1]; r[14] = iee[2]; r[15] = 0.0f;
}

// ---------------------------------------------------------------------------
// Per-(face,pixel) term: -softplus(±d²/σ). Inlined twice for 2-pixel ILP.
// ---------------------------------------------------------------------------
__device__ __forceinline__ float face_term(float gx, float gy,
                                           float4 e0, float4 e1, float4 e2,
                                           float4 ie) {
  float d2 = 3.402823466e+38f;
  bool allpos = true, allneg = true;
  {
    float pax = gx - e0.x, pay = gy - e0.y;
    float dote = pax * e0.z + pay * e0.w;
    float t = fminf(fmaxf(dote * ie.x, 0.0f), 1.0f);
    float dx = pax - t * e0.z, dy = pay - t * e0.w;
    d2 = fminf(d2, dx * dx + dy * dy);
    float crs = e0.z * pay - e0.w * pax;
    allpos = allpos && (crs >= 0.0f);
    allneg = allneg && (crs <= 0.0f);
  }
  {
    float pax = gx - e1.x, pay = gy - e1.y;
    float dote = pax * e1.z + pay * e1.w;
    float t = fminf(fmaxf(dote * ie.y, 0.0f), 1.0f);
    float dx = pax - t * e1.z, dy = pay - t * e1.w;
    d2 = fminf(d2, dx * dx + dy * dy);
    float crs = e1.z * pay - e1.w * pax;
    allpos = allpos && (crs >= 0.0f);
    allneg = allneg && (crs <= 0.0f);
  }
  {
    float pax = gx - e2.x, pay = gy - e2.y;
    float dote = pax * e2.z + pay * e2.w;
    float t = fminf(fmaxf(dote * ie.z, 0.0f), 1.0f);
    float dx = pax - t * e2.z, dy = pay - t * e2.w;
    d2 = fminf(d2, dx * dx + dy * dy);
    float crs = e2.z * pay - e2.w * pax;
    allpos = allpos && (crs >= 0.0f);
    allneg = allneg && (crs <= 0.0f);
  }
  bool inside = allpos || allneg;
  float z = (inside ? 1e5f : -1e5f) * d2;           // d2 / SIGMA
  // log_sigmoid(-z) = -softplus(z), numerically stable:
  return -(fmaxf(z, 0.0f) + log1pf(__expf(-fabsf(z))));
}

// ---------------------------------------------------------------------------
// Kernel 3: rasterize. grid = (nchunk, NPIX/512, B); 2 pixels per thread.
// Stage FCHUNK face records (16KB) into LDS via async global->LDS, then loop
// faces with broadcast ds_load_b128 reads. Disjoint accum slab per chunk.
// ---------------------------------------------------------------------------
__global__ __launch_bounds__(256) void raster_kernel(
    const float* __restrict__ rec, float* __restrict__ accum, int B, int F) {
  __shared__ float4 lrec[FCHUNK * 4];  // 16 KB

  const int tid   = threadIdx.x;
  const int chunk = blockIdx.x;
  const int tile  = blockIdx.y;
  const int b     = blockIdx.z;
  const int c0    = chunk * FCHUNK;
  const int nf    = min(FCHUNK, F - c0);

  // Stage nf*4 float4s, coalesced, 4 rounds of 256 lanes.
  const float4* gsrc = (const float4*)(rec + ((size_t)b * F + c0) * 16);
  const int nf4 = nf * 4;
#pragma unroll
  for (int j = 0; j < 4; ++j) {
    int idx = tid + j * 256;
    if (idx < nf4) stage_f4(gsrc, lrec, idx);
  }
  wait_async0();
  __syncthreads();

  const int p0 = tile * PXT + tid;        // this thread's two pixels
  const int p1 = p0 + 256;
  const float k1 = 1.0f / IMGW;
  const float gx0 =  (2.0f * (p0 % IMGW) + 1.0f - IMGW) * k1;
  const float gy0 = -(2.0f * (p0 / IMGW) + 1.0f - IMGW) * k1;
  const float gx1 =  (2.0f * (p1 % IMGW) + 1.0f - IMGW) * k1;
  const float gy1 = -(2.0f * (p1 / IMGW) + 1.0f - IMGW) * k1;

  float ssum0 = 0.0f, ssum1 = 0.0f;
  for (int f = 0; f < nf; ++f) {
    // All 32 lanes read the same LDS address -> conflict-free broadcast.
    float4 e0 = lrec[f * 4 + 0];
    float4 e1 = lrec[f * 4 + 1];
    float4 e2 = lrec[f * 4 + 2];
    float4 ie = lrec[f * 4 + 3];
    ssum0 += face_term(gx0, gy0, e0, e1, e2, ie);
    ssum1 += face_term(gx1, gy1, e0, e1, e2, ie);
  }

  float* slab = accum + (((size_t)chunk * B) + b) * NPIX;
  slab[p0] = ssum0;
  slab[p1] = ssum1;
}

// ---------------------------------------------------------------------------
// Kernel 4: alpha = 1 - exp(sum over chunks), fixed order. grid=(NPIX/256, B).
// ---------------------------------------------------------------------------
__global__ __launch_bounds__(256) void final_kernel(
    const float* __restrict__ accum, float* __restrict__ out, int B, int nchunk) {
  const int p = blockIdx.x * blockDim.x + threadIdx.x;
  if (p >= NPIX) return;
  const int b = blockIdx.y;
  float s = 0.0f;
  for (int c = 0; c < nchunk; ++c)
    s += accum[(((size_t)c * B) + b) * NPIX + p];
  out[(size_t)b * NPIX + p] = 1.0f - expf(s);
}

extern "C" void kernel_launch(void* const* d_in, const int* in_sizes, int n_in,
                              void* d_out, int out_size, void* d_ws, size_t ws_size,
                              hipStream_t stream) {
  const float* verts = (const float*)d_in[0];
  const int*   faces = (const int*)d_in[1];
  const float* cams  = (const float*)d_in[2];

  const int B = in_sizes[2] / 7;                 // cams (B,7)
  const int V = in_sizes[0] / (3 * B);           // vertices (B,V,3)
  const int F = in_sizes[1] / (3 * B);           // faces (B,F,3)
  const int nchunk = (F + FCHUNK - 1) / FCHUNK;

  // Workspace layout (floats), 16B-aligned sections:
  size_t xy_elems = ((size_t)B * V * 2 + 3) & ~(size_t)3;
  float* xy    = (float*)d_ws;                   // B*V*2
  float* rec   = xy + xy_elems;                  // B*F*16
  float* accum = rec + (size_t)B * F * 16;       // nchunk*B*NPIX

  {
    dim3 grid((V + 255) / 256, B);
    vert_kernel<<<grid, 256, 0, stream>>>(verts, cams, xy, V);
  }
  {
    dim3 grid((F + 255) / 256, B);
    face_kernel<<<grid, 256, 0, stream>>>(faces, xy, rec, V, F);
  }
  {
    dim3 grid(nchunk, NPIX / PXT, B);            // (5, 18, 2)
    raster_kernel<<<grid, 256, 0, stream>>>(rec, accum, B, F);
  }
  {
    dim3 grid((NPIX + 255) / 256, B);
    final_kernel<<<grid, 256, 0, stream>>>(accum, (float*)d_out, B, nchunk);
  }
  (void)n_in; (void)out_size; (void)ws_size;
}